// MHAttention_83820581749487
// MI455X (gfx1250) — compile-verified
//
#include <hip/hip_runtime.h>
#include <hip/hip_bf16.h>

// MHA forward for MI455X (gfx1250), fp32 end-to-end on the WMMA f32 pipe.
// Logits ~ N(0, 64^2) -> softmax is near-one-hot; bf16/f16 score math would
// corrupt it. ~60 GFLOP vs ~180MB traffic @ 23.3TB/s -> memory-balanced, so
// exact-f32 V_WMMA_F32_16X16X4_F32 costs nothing on the roofline.
// Tile staging rides the Tensor Data Mover (tensor_load_to_lds +
// s_wait_tensorcnt, double-buffered); TDM pad fields generate the LDS
// bank-padding strides.

typedef __attribute__((ext_vector_type(2))) float        v2f;
typedef __attribute__((ext_vector_type(8))) float        v8f;
typedef __attribute__((ext_vector_type(4))) unsigned int v4u;
typedef __attribute__((ext_vector_type(8))) int          v8i;
typedef __attribute__((ext_vector_type(4))) int          v4i;

#define S_LEN  2048
#define BATCH  2
#define DMODEL 1024
#define NHEAD  16
#define DKH    64

static __device__ __forceinline__ v8f wmma_f32(v2f a, v2f b, v8f c) {
  // D = A(16x4) * B(4x16) + C(16x16), all f32:
  // (neg_a, A, neg_b, B, c_mod, C, reuse_a, reuse_b)
  return __builtin_amdgcn_wmma_f32_16x16x4_f32(false, a, false, b, (short)0, c,
                                               false, false);
}

static __device__ __forceinline__ unsigned lds_off(const void* p) {
  // Generic (flat) pointers to LDS carry the wave-relative LDS byte address in
  // their low 32 bits (ISA 10.2 aperture mapping).
  return (unsigned)(unsigned long long)p;
}

// TDM 2D tile load, 6-arg builtin (this toolchain: clang-23/therock-10.0:
// (v4u g0, v8i g1, v4i g2, v4i g3, v8i extra, i32 cpol)).
// D# layout per ISA 08 §8.3/8.4:
//  group0: count=1 | lds_addr | global_addr[56:0] | type=2
//  group1: data_size=4B, pad_enable, pad_interval/amount, tensor dims = tile
//          dims (no OOB), tensor_dim0_stride = row stride in elements.
//  groups 2/3 zero (<=2D tile), cpol 0.
// padIntervalCode: pad after 2<<code DWORDs; padAmountCode: (code+1) DWORDs.
static __device__ __forceinline__ void tdm_load_2d(const float* gsrc,
                                                   unsigned ldsByteOff,
                                                   unsigned tileW, unsigned tileH,
                                                   unsigned rowStrideElems,
                                                   unsigned padIntervalCode,
                                                   unsigned padAmountCode) {
  const unsigned long long ga = (unsigned long long)gsrc;
  v4u g0;
  g0[0] = 1u;                                            // count=1, user desc
  g0[1] = ldsByteOff;                                    // lds_addr
  g0[2] = (unsigned)ga;                                  // global_addr[31:0]
  g0[3] = (unsigned)((ga >> 32) & 0x01FFFFFFu) | (2u << 30);  // ga[56:32]|type=2
  v8i g1;
  g1[0] = (int)((2u << 16) | (1u << 20) |                // data_size=4B, pad_en
                (padIntervalCode << 22) | (padAmountCode << 25));
  g1[1] = (int)((tileW & 0xFFFFu) << 16);                // tensor_dim0[15:0]
  g1[2] = (int)((tileW >> 16) | ((tileH & 0xFFFFu) << 16));   // |tensor_dim1 lo
  g1[3] = (int)((tileH >> 16) | (tileW << 16));          // | tile_dim0
  g1[4] = (int)tileH;                                    // tile_dim1 (dim2=0)
  g1[5] = (int)rowStrideElems;                           // dim0_stride[31:0]
  g1[6] = 0;                                             // dim0_stride[47:32]
  g1[7] = 0;
  const v4i z4 = {0, 0, 0, 0};
  const v8i z8 = {0, 0, 0, 0, 0, 0, 0, 0};
  __builtin_amdgcn_tensor_load_to_lds(g0, g1, z4, z4, z8, 0);
}

// ---------------------------------------------------------------------------
// Projection: Y[M=4096, N=1024] = X[M,1024] @ W[N,1024]^T (+ bias)
// 8 waves (4x2) -> 64x32 tile, K chunked by 32. X/W tiles double-buffered via
// TDM (wave 0 drives DMA, s_wait_tensorcnt(2) = previous pair landed).
// A-frag (16x4 f32): lane<16 -> row M=lane, {K,K+1}; lanes 16..31 -> {K+2,K+3}.
// B-frag mirrors with lanes indexing N; B[k][n]=W[n][k] -> contiguous rows.
// ---------------------------------------------------------------------------
#define PK     1024
#define KC     32
#define NKT    (PK / KC)
#define XS_STR 36   // 32 floats + 4 pad (TDM pad_interval=4 (32 DW), amount=3 (4 DW))
#define WS_STR 36

__global__ __launch_bounds__(256) void proj_kernel(const float* __restrict__ X,
                                                   const float* __restrict__ W,
                                                   const float* __restrict__ bias,
                                                   float* __restrict__ Y) {
  __shared__ __align__(16) float Xs[2][64 * XS_STR];
  __shared__ __align__(16) float Ws[2][32 * WS_STR];
  const int t    = threadIdx.x;
  const int lane = t & 31, wave = t >> 5;
  const int wr   = wave & 3, wc = wave >> 2;       // wave grid 4 (M) x 2 (N)
  const int l15  = lane & 15, hi = lane >> 4;
  const int M0   = blockIdx.y * 64, N0 = blockIdx.x * 32;

  if (wave == 0) {                                 // prologue: tile 0 -> buf 0
    tdm_load_2d(&X[(size_t)M0 * PK], lds_off(&Xs[0][0]), KC, 64, PK, 4, 3);
    tdm_load_2d(&W[(size_t)N0 * PK], lds_off(&Ws[0][0]), KC, 32, PK, 4, 3);
  }

  v8f acc = {};
  for (int it = 0; it < NKT; ++it) {
    if (wave == 0) {
      const int nx = it + 1;
      if (nx < NKT) {                              // kick tile it+1 -> other buf
        tdm_load_2d(&X[(size_t)M0 * PK + nx * KC], lds_off(&Xs[nx & 1][0]),
                    KC, 64, PK, 4, 3);
        tdm_load_2d(&W[(size_t)N0 * PK + nx * KC], lds_off(&Ws[nx & 1][0]),
                    KC, 32, PK, 4, 3);
        __builtin_amdgcn_s_wait_tensorcnt(2);      // in-order TDM: tile it done
      } else {
        __builtin_amdgcn_s_wait_tensorcnt(0);
      }
    }
    __syncthreads();                               // tile it visible to all

    const float* Xc = &Xs[it & 1][0];
    const float* Wc = &Ws[it & 1][0];
#pragma unroll
    for (int kk4 = 0; kk4 < KC / 4; ++kk4) {
      const int kOff = kk4 * 4 + 2 * hi;
      const v2f a = *(const v2f*)&Xc[(wr * 16 + l15) * XS_STR + kOff];
      const v2f b = *(const v2f*)&Wc[(wc * 16 + l15) * WS_STR + kOff];
      acc = wmma_f32(a, b, acc);
    }
    __syncthreads();                               // done reading buf it&1
  }

  const float bv = bias ? bias[N0 + wc * 16 + l15] : 0.0f;
#pragma unroll
  for (int v = 0; v < 8; ++v) {                    // C layout: row v+8*hi, col l15
    const int m = M0 + wr * 16 + v + 8 * hi;
    Y[(size_t)m * DMODEL + N0 + wc * 16 + l15] = acc[v] + bv;
  }
}

// ---------------------------------------------------------------------------
// Flash attention: wave = 16 query rows (Q frags resident in VGPRs), block =
// 8 waves = 128 rows of one (b,h). K/V 32x64 tiles double-buffered via TDM.
// Online softmax with 16-lane shuffle reductions over the C-layout tiles
// (lane&15 = column N, VGPR v = row v+8*hi); P bounced through wave-private
// LDS to rebuild A-frags for the P·V WMMAs.
// ---------------------------------------------------------------------------
#define JC     32
#define NJT    (S_LEN / JC)
#define KS_STR 68   // 64 floats + 4 pad (TDM pad_interval=5 (64 DW), amount=3)
#define PS_STR 36

__global__ __launch_bounds__(256) void attn_kernel(const float* __restrict__ Qm,
                                                   const float* __restrict__ Km,
                                                   const float* __restrict__ Vm,
                                                   float* __restrict__ Out) {
  __shared__ __align__(16) float Ks[2][JC * KS_STR];
  __shared__ __align__(16) float Vs[2][JC * KS_STR];
  __shared__ __align__(16) float Ps[8][16 * PS_STR];
  const int t    = threadIdx.x;
  const int lane = t & 31, wave = t >> 5;
  const int l15  = lane & 15, hi = lane >> 4;
  const int b    = blockIdx.y & (BATCH - 1);
  const int h    = blockIdx.y >> 1;
  const int i0   = blockIdx.x * 128 + wave * 16;
  const long hd  = (long)h * DKH;
  const long kvRow = (long)BATCH * DMODEL;         // stride between j rows

  if (wave == 0) {                                 // prologue: j-tile 0 -> buf 0
    const float* kp = &Km[(long)b * DMODEL + hd];
    const float* vp = &Vm[(long)b * DMODEL + hd];
    tdm_load_2d(kp, lds_off(&Ks[0][0]), DKH, JC, (unsigned)kvRow, 5, 3);
    tdm_load_2d(vp, lds_off(&Vs[0][0]), DKH, JC, (unsigned)kvRow, 5, 3);
  }

  // Q A-fragments for this wave's 16 rows: 16 x (dk=64) -> 16 v2f in VGPRs.
  v2f qf[16];
  {
    const long rowBase = ((long)(i0 + l15) * BATCH + b) * DMODEL + hd;
#pragma unroll
    for (int c = 0; c < 16; ++c)
      qf[c] = *(const v2f*)&Qm[rowBase + c * 4 + 2 * hi];
  }

  v8f o0 = {}, o1 = {}, o2 = {}, o3 = {};
  float rmax[8], rsum[8];
#pragma unroll
  for (int v = 0; v < 8; ++v) { rmax[v] = -3.0e38f; rsum[v] = 0.0f; }

  for (int it = 0; it < NJT; ++it) {
    if (wave == 0) {
      const int nx = it + 1;
      if (nx < NJT) {                              // kick j-tile it+1
        const long gj = ((long)(nx * JC) * BATCH + b) * DMODEL + hd;
        tdm_load_2d(&Km[gj], lds_off(&Ks[nx & 1][0]), DKH, JC, (unsigned)kvRow, 5, 3);
        tdm_load_2d(&Vm[gj], lds_off(&Vs[nx & 1][0]), DKH, JC, (unsigned)kvRow, 5, 3);
        __builtin_amdgcn_s_wait_tensorcnt(2);      // tile it landed
      } else {
        __builtin_amdgcn_s_wait_tensorcnt(0);
      }
    }
    __syncthreads();

    const float* Kc = &Ks[it & 1][0];
    const float* Vc = &Vs[it & 1][0];

    // Scores S[16 x 32] = Q * K^T  (B[k][n] = K[j0+n][k] -> contiguous rows)
    v8f s0 = {}, s1 = {};
#pragma unroll
    for (int c = 0; c < 16; ++c) {
      const int kOff = c * 4 + 2 * hi;
      const v2f b0 = *(const v2f*)&Kc[l15 * KS_STR + kOff];
      const v2f b1 = *(const v2f*)&Kc[(16 + l15) * KS_STR + kOff];
      s0 = wmma_f32(qf[c], b0, s0);
      s1 = wmma_f32(qf[c], b1, s1);
    }

    // Online softmax; row owned by (v, half): M = v + 8*hi.
#pragma unroll
    for (int v = 0; v < 8; ++v) {
      const float x0 = s0[v] * 8.0f;               // * sqrt(dk), per reference
      const float x1 = s1[v] * 8.0f;
      float mx = fmaxf(x0, x1);
#pragma unroll
      for (int off = 1; off < 16; off <<= 1)
        mx = fmaxf(mx, __shfl_xor(mx, off, 32));
      const float mnew  = fmaxf(rmax[v], mx);
      const float alpha = __expf(rmax[v] - mnew);
      const float p0 = __expf(x0 - mnew);
      const float p1 = __expf(x1 - mnew);
      float ps = p0 + p1;
#pragma unroll
      for (int off = 1; off < 16; off <<= 1)
        ps += __shfl_xor(ps, off, 32);
      rsum[v] = rsum[v] * alpha + ps;
      rmax[v] = mnew;
      o0[v] *= alpha; o1[v] *= alpha; o2[v] *= alpha; o3[v] *= alpha;
      Ps[wave][(v + 8 * hi) * PS_STR + l15]      = p0;   // C layout -> LDS
      Ps[wave][(v + 8 * hi) * PS_STR + 16 + l15] = p1;
    }
    // Wave-private LDS bounce; in-wave DS ordering + auto s_wait_dscnt suffice.

    // O[16x64] += P[16x32] * V[32x64]
#pragma unroll
    for (int kc = 0; kc < 8; ++kc) {
      const int kOff = kc * 4 + 2 * hi;
      const v2f a = *(const v2f*)&Ps[wave][l15 * PS_STR + kOff];
      v2f b0, b1, b2, b3;
      b0.x = Vc[(kOff + 0) * KS_STR + l15];      b0.y = Vc[(kOff + 1) * KS_STR + l15];
      b1.x = Vc[(kOff + 0) * KS_STR + 16 + l15]; b1.y = Vc[(kOff + 1) * KS_STR + 16 + l15];
      b2.x = Vc[(kOff + 0) * KS_STR + 32 + l15]; b2.y = Vc[(kOff + 1) * KS_STR + 32 + l15];
      b3.x = Vc[(kOff + 0) * KS_STR + 48 + l15]; b3.y = Vc[(kOff + 1) * KS_STR + 48 + l15];
      o0 = wmma_f32(a, b0, o0);
      o1 = wmma_f32(a, b1, o1);
      o2 = wmma_f32(a, b2, o2);
      o3 = wmma_f32(a, b3, o3);
    }
    __syncthreads();                               // done reading buf it&1
  }

  // Normalize and store [S,B,D] output.
#pragma unroll
  for (int v = 0; v < 8; ++v) {
    const float inv = 1.0f / rsum[v];
    const long g = ((long)(i0 + v + 8 * hi) * BATCH + b) * DMODEL + hd;
    Out[g + l15]      = o0[v] * inv;
    Out[g + 16 + l15] = o1[v] * inv;
    Out[g + 32 + l15] = o2[v] * inv;
    Out[g + 48 + l15] = o3[v] * inv;
  }
}

// ---------------------------------------------------------------------------
extern "C" void kernel_launch(void* const* d_in, const int* in_sizes, int n_in,
                              void* d_out, int out_size, void* d_ws, size_t ws_size,
                              hipStream_t stream) {
  const float* query = (const float*)d_in[0];
  const float* key   = (const float*)d_in[1];
  const float* value = (const float*)d_in[2];
  const float* Wq    = (const float*)d_in[3];
  const float* Wk    = (const float*)d_in[4];
  const float* Wv    = (const float*)d_in[5];
  const float* bv    = (const float*)d_in[6];

  const size_t mat = (size_t)S_LEN * BATCH * DMODEL;   // 4096*1024 floats
  float* Qp = (float*)d_ws;                            // needs 3*mat*4 = 48 MiB
  float* Kp = Qp + mat;
  float* Vp = Kp + mat;

  dim3 blk(256);
  dim3 pg(DMODEL / 32, (S_LEN * BATCH) / 64);          // 32 x 64 blocks
  proj_kernel<<<pg, blk, 0, stream>>>(query, Wq, nullptr, Qp);
  proj_kernel<<<pg, blk, 0, stream>>>(key,   Wk, nullptr, Kp);
  proj_kernel<<<pg, blk, 0, stream>>>(value, Wv, bv,      Vp);

  dim3 ag(S_LEN / 128, BATCH * NHEAD);                 // 16 x 32 blocks
  attn_kernel<<<ag, blk, 0, stream>>>(Qp, Kp, Vp, (float*)d_out);
}